// GCN_18708877541972
// MI455X (gfx1250) — compile-verified
//
#include <hip/hip_runtime.h>
#include <stdint.h>

typedef __attribute__((ext_vector_type(2))) float v2f;
typedef __attribute__((ext_vector_type(8))) float v8f;

// ---------------------------------------------------------------------------
// fill: p[i] = v for i < n
// ---------------------------------------------------------------------------
__global__ void fill_kernel(float* __restrict__ p, float v, int n) {
    int i = blockIdx.x * blockDim.x + threadIdx.x;
    if (i < n) p[i] = v;
}

// ---------------------------------------------------------------------------
// degree accumulation: deg[dst] += 1.0 for every edge (deg pre-filled to 2.0)
// edge_index layout: [2, E] row-major int64 -> src = ei[e], dst = ei[E + e]
// ---------------------------------------------------------------------------
__global__ void deg_kernel(const long long* __restrict__ ei, float* __restrict__ deg, int E) {
    int e = blockIdx.x * blockDim.x + threadIdx.x;
    if (e >= E) return;
    int d = (int)ei[E + e];
    atomicAdd(&deg[d], 1.0f);
}

// in-place: deg[i] -> rsqrt(deg[i])   (deg >= 2 always, no zero guard needed)
__global__ void rsqrt_kernel(float* __restrict__ deg, int n) {
    int i = blockIdx.x * blockDim.x + threadIdx.x;
    if (i < n) deg[i] = rsqrtf(deg[i]);
}

// ---------------------------------------------------------------------------
// H[M,Nout] = X[M,K] @ W[K,Nout] via V_WMMA_F32_16X16X4_F32, one wave per
// 16x16 output tile.  wave32; 32-bit A 16x4 layout: lanes 0-15 carry K={0,1}
// in v[0],v[1]; lanes 16-31 carry K={2,3}.  B (4x16) mirrored.  D: VGPR j ->
// rows j (lanes 0-15) and j+8 (lanes 16-31), col = lane&15.
// M must be a multiple of 16; K,Nout multiples of 4/16 (64,64 / 64,32 here).
// ---------------------------------------------------------------------------
__global__ __launch_bounds__(256) void gemm_wmma_kernel(
        const float* __restrict__ X, const float* __restrict__ W,
        float* __restrict__ H, int M, int K, int Nout) {
    int wave   = (blockIdx.x * blockDim.x + threadIdx.x) >> 5;
    int lane   = threadIdx.x & 31;
    int tilesN = Nout >> 4;
    int tm     = wave / tilesN;
    int tn     = wave - tm * tilesN;
    if (tm * 16 >= M) return;           // uniform per wave -> EXEC stays all-1s

    int laneHi = lane >> 4;             // 0 or 1
    int lane15 = lane & 15;
    int row    = tm * 16 + lane15;      // A row owned by this lane
    int col    = tn * 16 + lane15;      // B/D column owned by this lane

    const float* xr = X + (long long)row * K;
    v8f c = {};
    for (int k0 = 0; k0 < K; k0 += 4) {
        int ka = k0 + (laneHi << 1);
        v2f a, b;
        a.x = xr[ka];
        a.y = xr[ka + 1];
        b.x = W[(long long)ka * Nout + col];
        b.y = W[(long long)(ka + 1) * Nout + col];
        // 8 args: (neg_a, A, neg_b, B, c_mod, C, reuse_a, reuse_b)
        c = __builtin_amdgcn_wmma_f32_16x16x4_f32(
                false, a, false, b, (short)0, c, false, false);
    }

    int rbase = tm * 16 + laneHi * 8;
#pragma unroll
    for (int j = 0; j < 8; ++j) {
        H[(long long)(rbase + j) * Nout + col] = c[j];
    }
}

// ---------------------------------------------------------------------------
// agg[dst,:] += h[src,:] * (dinv[src]*dinv[dst])  over all edges.
// gid >> chunkBits = edge; low bits select a float4 feature chunk, so the
// F/4 lanes of an edge issue one coalesced 16B-per-lane gather of h[src,:].
// ---------------------------------------------------------------------------
__global__ void edge_agg_kernel(const long long* __restrict__ ei,
                                const float* __restrict__ h,
                                const float* __restrict__ dinv,
                                float* __restrict__ agg,
                                int E, int F, int chunkBits) {
    long long gid = (long long)blockIdx.x * blockDim.x + threadIdx.x;
    int e = (int)(gid >> chunkBits);
    if (e >= E) return;
    int c = ((int)gid & ((1 << chunkBits) - 1)) << 2;

    int s = (int)ei[e];
    int d = (int)ei[E + e];
    float w = dinv[s] * dinv[d];

    const float4 hv = *(const float4*)(h + (long long)s * F + c);
    float* ap = agg + (long long)d * F + c;
    atomicAdd(ap + 0, hv.x * w);
    atomicAdd(ap + 1, hv.y * w);
    atomicAdd(ap + 2, hv.z * w);
    atomicAdd(ap + 3, hv.w * w);
}

// ---------------------------------------------------------------------------
// out[i,f] = agg[i,f] + h[i,f] * (2*dinv[i]^2) + bias[f]   (+ optional ReLU)
// ---------------------------------------------------------------------------
__global__ void post_kernel(const float* __restrict__ agg,
                            const float* __restrict__ h,
                            const float* __restrict__ dinv,
                            const float* __restrict__ bias,
                            float* __restrict__ out,
                            int nfeat, int n, int do_relu) {
    long long gid = (long long)blockIdx.x * blockDim.x + threadIdx.x;
    long long total = (long long)n * nfeat;
    if (gid >= total) return;
    int i = (int)(gid / nfeat);
    int f = (int)(gid - (long long)i * nfeat);
    float di = dinv[i];
    float v = agg[gid] + h[gid] * (2.0f * di * di) + bias[f];
    if (do_relu) v = fmaxf(v, 0.0f);
    out[gid] = v;
}

// ---------------------------------------------------------------------------
// launch
// ---------------------------------------------------------------------------
extern "C" void kernel_launch(void* const* d_in, const int* in_sizes, int n_in,
                              void* d_out, int out_size, void* d_ws, size_t ws_size,
                              hipStream_t stream) {
    const float*     x  = (const float*)d_in[0];
    const long long* ei = (const long long*)d_in[1];   // int64 edge_index [2,E]
    const float*     W1 = (const float*)d_in[2];
    const float*     b1 = (const float*)d_in[3];
    const float*     W2 = (const float*)d_in[4];
    const float*     b2 = (const float*)d_in[5];

    const int N = in_sizes[0] / 64;     // 100000
    const int E = in_sizes[1] / 2;      // 1600000
    const int F1 = 64, F2 = 32;

    // workspace layout (floats)
    float* deg  = (float*)d_ws;               // [N]   -> becomes dinv in place
    float* h1   = deg  + N;                   // [N*64]
    float* agg1 = h1   + (size_t)N * F1;      // [N*64]
    float* h2   = agg1 + (size_t)N * F1;      // [N*32]
    float* agg2 = h2   + (size_t)N * F2;      // [N*32]

    // output layout: (out, feature_map) concatenated
    float* outp = (float*)d_out;              // [N*32]
    float* fm   = outp + (size_t)N * F2;      // [N*64]

    const int T = 256;
    auto blocks = [](long long n, int t) { return (unsigned)((n + t - 1) / t); };

    // 1) init: deg = 2.0 (improved self-loop weight), agg buffers = 0
    fill_kernel<<<blocks(N, T), T, 0, stream>>>(deg, 2.0f, N);
    fill_kernel<<<blocks((long long)N * F1, T), T, 0, stream>>>(agg1, 0.0f, N * F1);
    fill_kernel<<<blocks((long long)N * F2, T), T, 0, stream>>>(agg2, 0.0f, N * F2);

    // 2) degrees + dinv = rsqrt(deg)
    deg_kernel<<<blocks(E, T), T, 0, stream>>>(ei, deg, E);
    rsqrt_kernel<<<blocks(N, T), T, 0, stream>>>(deg, N);
    const float* dinv = deg;

    // 3) layer 1: h1 = x @ W1  (WMMA, one wave per 16x16 tile)
    {
        long long waves = (long long)(N / 16) * (F1 / 16);
        gemm_wmma_kernel<<<blocks(waves * 32, T), T, 0, stream>>>(x, W1, h1, N, 64, F1);
    }
    edge_agg_kernel<<<blocks((long long)E << 4, T), T, 0, stream>>>(ei, h1, dinv, agg1, E, F1, 4);
    post_kernel<<<blocks((long long)N * F1, T), T, 0, stream>>>(agg1, h1, dinv, b1, fm, F1, N, 1);

    // 4) layer 2: h2 = fm @ W2
    {
        long long waves = (long long)(N / 16) * (F2 / 16);
        gemm_wmma_kernel<<<blocks(waves * 32, T), T, 0, stream>>>(fm, W2, h2, N, 64, F2);
    }
    edge_agg_kernel<<<blocks((long long)E << 3, T), T, 0, stream>>>(ei, h2, dinv, agg2, E, F2, 3);
    post_kernel<<<blocks((long long)N * F2, T), T, 0, stream>>>(agg2, h2, dinv, b2, outp, F2, N, 0);
}